// OperatorModule_36713380446469
// MI455X (gfx1250) — compile-verified
//
#include <hip/hip_runtime.h>
#include <hip/hip_bf16.h>
#include <math.h>

typedef __attribute__((ext_vector_type(2))) float v2f;
typedef __attribute__((ext_vector_type(8))) float v8f;

#define NVOXC   256
#define NUC     128
#define NVC     128
#define NANGC   32
#define DSO_F   1000.0f
#define DSD_F   1536.0f
#define DU_F    3.0f
#define RAD_F   221.70250336881628f   /* 256*sqrt(3)/2 */
#define STEP_F  1.7320508075688772f   /* 2*RAD/256 : constant sample spacing */

__global__ __launch_bounds__(256) void cone_fp_wmma_kernel(
    const float* __restrict__ vol, float* __restrict__ out) {
  const int lane   = threadIdx.x & 31;
  const int m      = lane & 15;   // matrix row = ray-in-wave
  const int h      = lane >> 4;   // K half: lanes 16..31 supply K=2,3
  const int waveId = threadIdx.x >> 5;
  const int rayBase = (blockIdx.x * 8 + waveId) * 16;
  const int ray     = rayBase + m;

  const int u = ray & (NUC - 1);
  const int v = (ray >> 7) & (NVC - 1);
  const int a = ray >> 14;

  // ---- per-ray geometry (matches reference exactly) ----
  const float theta = (float)a * (6.2831853071795864f / 32.0f);
  float st, ct;
  sincosf(theta, &st, &ct);
  const float us = ((float)u - 63.5f) * DU_F;
  const float vs = ((float)v - 63.5f) * DU_F;
  // ray = pix - src ; src = (DSO*ct, DSO*st, 0)
  const float rx = -DSD_F * ct - st * us;
  const float ry = -DSD_F * st + ct * us;
  const float rz = vs;
  const float rlen = sqrtf(rx * rx + ry * ry + rz * rz);
  const float inv  = 1.0f / rlen;
  const float t0   = (DSO_F - RAD_F) * inv;
  const float dts  = (2.0f * RAD_F * inv) * (1.0f / 256.0f); // dt per sample
  const float tA   = t0 + 0.5f * dts;                        // t at sample 0
  // voxel-index-space start and step: f = pos + 127.5 (DVOX = 1mm)
  const float f0x = fmaf(rx, tA, DSO_F * ct) + 127.5f;
  const float f0y = fmaf(ry, tA, DSO_F * st) + 127.5f;
  const float f0z = rz * tA + 127.5f;
  const float dvx = rx * dts, dvy = ry * dts, dvz = rz * dts;

  // ---- branch-light trilinear sampler (zero outside volume) ----
  auto sample = [&](float sf) -> float {
    const float fx = fmaf(sf, dvx, f0x);
    const float fy = fmaf(sf, dvy, f0y);
    const float fz = fmaf(sf, dvz, f0z);
    float val = 0.0f;
    if (fx > -1.0f && fx < 256.0f && fy > -1.0f && fy < 256.0f &&
        fz > -1.0f && fz < 256.0f) {
      const float xf = floorf(fx), yf = floorf(fy), zf = floorf(fz);
      const int ix = (int)xf, iy = (int)yf, iz = (int)zf;
      const float wx = fx - xf, wy = fy - yf, wz = fz - zf;
      // fold per-corner validity into per-axis weights (clamped addresses)
      const float wx1 = (ix <= 254) ? wx : 0.0f;
      const float wx0 = (ix >= 0) ? 1.0f - wx : 0.0f;
      const float wy1 = (iy <= 254) ? wy : 0.0f;
      const float wy0 = (iy >= 0) ? 1.0f - wy : 0.0f;
      const float wz1 = (iz <= 254) ? wz : 0.0f;
      const float wz0 = (iz >= 0) ? 1.0f - wz : 0.0f;
      const int ox0 = (ix < 0) ? 0 : ix;
      const int ox1 = (ix >= 255) ? 255 : ix + 1;
      const int oy0 = ((iy < 0) ? 0 : iy) << 8;
      const int oy1 = ((iy >= 255) ? 255 : iy + 1) << 8;
      const int oz0 = ((iz < 0) ? 0 : iz) << 16;
      const int oz1 = ((iz >= 255) ? 255 : iz + 1) << 16;
      const float v000 = vol[oz0 + oy0 + ox0];
      const float v100 = vol[oz0 + oy0 + ox1];
      const float v010 = vol[oz0 + oy1 + ox0];
      const float v110 = vol[oz0 + oy1 + ox1];
      const float v001 = vol[oz1 + oy0 + ox0];
      const float v101 = vol[oz1 + oy0 + ox1];
      const float v011 = vol[oz1 + oy1 + ox0];
      const float v111 = vol[oz1 + oy1 + ox1];
      const float c00 = fmaf(v100, wx1, v000 * wx0);
      const float c10 = fmaf(v110, wx1, v010 * wx0);
      const float c01 = fmaf(v101, wx1, v001 * wx0);
      const float c11 = fmaf(v111, wx1, v011 * wx0);
      const float c0 = fmaf(c10, wy1, c00 * wy0);
      const float c1 = fmaf(c11, wy1, c01 * wy0);
      val = fmaf(c1, wz1, c0 * wz0);
    }
    return val;
  };

  // ---- march 256 samples; WMMA (B = ones) performs the K-reduction ----
  v8f acc = {0.f, 0.f, 0.f, 0.f, 0.f, 0.f, 0.f, 0.f};
  const v2f bones = {1.0f, 1.0f};
  const float kOff = (float)(2 * h); // K offset this half-wave supplies

#pragma unroll 2
  for (int c = 0; c < 64; ++c) {
    const float s0 = (float)(4 * c) + kOff;
    v2f amat;
    amat.x = sample(s0);
    amat.y = sample(s0 + 1.0f);

    // prefetch the base voxel line of this lane's sample two chunks ahead
    {
      const float sp = (float)(4 * (c + 2)) + kOff;
      const float pfx = fmaf(sp, dvx, f0x);
      const float pfy = fmaf(sp, dvy, f0y);
      const float pfz = fmaf(sp, dvz, f0z);
      int px = (int)pfx, py = (int)pfy, pz = (int)pfz;
      px = (px < 0) ? 0 : ((px > 255) ? 255 : px);
      py = (py < 0) ? 0 : ((py > 255) ? 255 : py);
      pz = (pz < 0) ? 0 : ((pz > 255) ? 255 : pz);
      __builtin_prefetch(vol + ((pz << 16) + (py << 8) + px), 0, 3);
    }

    // D[m][n] += sum_k A[m,k]*1 : row-sum accumulate on the matrix pipe
    acc = __builtin_amdgcn_wmma_f32_16x16x4_f32(
        false, amat, false, bones, (short)0, acc, false, false);
  }

  // ---- column N=0 (lanes 0 and 16) holds rows 0-7 / 8-15 of D ----
  if ((lane & 15) == 0) {
    const int a2 = rayBase >> 14;
    const int v2 = (rayBase >> 7) & (NVC - 1);
    const int u0 = rayBase & (NUC - 1);
    float* orow = out + ((size_t)v2 * NANGC + a2) * NUC + u0;
    const int mb = h * 8;
#pragma unroll
    for (int g = 0; g < 8; ++g) {
      orow[mb + g] = acc[g] * STEP_F;
    }
  }
}

extern "C" void kernel_launch(void* const* d_in, const int* in_sizes, int n_in,
                              void* d_out, int out_size, void* d_ws, size_t ws_size,
                              hipStream_t stream) {
  const float* vol = (const float*)d_in[0];
  float* out = (float*)d_out;
  // 524288 rays total; 128 rays per 256-thread block (8 waves x 16 rays)
  dim3 grid(4096), block(256);
  hipLaunchKernelGGL(cone_fp_wmma_kernel, grid, block, 0, stream, vol, out);
  (void)in_sizes; (void)n_in; (void)out_size; (void)d_ws; (void)ws_size;
}